// Conv1dWithMeanAndThresholdChannels_36412732736195
// MI455X (gfx1250) — compile-verified
//
#include <hip/hip_runtime.h>

typedef __attribute__((ext_vector_type(2))) float v2f;
typedef __attribute__((ext_vector_type(8))) float v8f;

#define OUTC    64
#define KSZ     19
#define KPAD    20      // pad K to 5 WMMA steps of 4 (column 19 of sW is zero)
#define CSTRIDE 5
#define PADV    9
#define LIN     100000
#define LOUT    20000
#define UPPERV  2.99f
#define LOWERV  (-2.99f)

#define WAVES_PER_BLOCK 8
#define TILE_N          16
#define BLOCK_T         (WAVES_PER_BLOCK * TILE_N)          // 128 output positions / block
#define XS_LEN          ((BLOCK_T - 1) * CSTRIDE + KSZ + 2) // 654 valid + pad slots (all staged)
#define BLOCKS_PER_BATCH 157                                 // ceil(1250 tiles / 8)

__global__ __launch_bounds__(256)
void Conv1dWithMeanAndThresholdChannels_kernel(const float* __restrict__ x,
                                               const float* __restrict__ W,
                                               const float* __restrict__ bias,
                                               float* __restrict__ out)
{
    __shared__ float sW[OUTC * KPAD];   // 64 x 20 weight matrix (row0 = mean, row1 = 0)
    __shared__ float sx[XS_LEN];        // contiguous x window for this block (zero padded)
    __shared__ float sb[OUTC];          // bias (0 for ch 0,1)

    const int tid = threadIdx.x;
    const int bb  = blockIdx.x / BLOCKS_PER_BATCH;
    const int blk = blockIdx.x % BLOCKS_PER_BATCH;
    const int T0  = blk * BLOCK_T;                  // first output position handled by block

    // ---- stage weights: row0 = 1/K (mean), row1 = 0 (threshold slot), rows 2..63 = W
    for (int i = tid; i < OUTC * KPAD; i += 256) {
        const int row = i / KPAD, k = i % KPAD;
        float v = 0.0f;
        if (k < KSZ) {
            if (row == 0)      v = 1.0f / (float)KSZ;
            else if (row >= 2) v = W[(row - 2) * KSZ + k];
        }
        sW[i] = v;
    }
    if (tid < OUTC) sb[tid] = (tid < 2) ? 0.0f : bias[tid - 2];

    // ---- stage x window: global indices [T0*5-9, T0*5-9+XS_LEN)
    const long lo = (long)T0 * CSTRIDE - PADV;
    const float* xb = x + (long)bb * LIN;
    for (int i = tid; i < XS_LEN; i += 256) {
        const long gi = lo + i;
        sx[i] = (gi >= 0 && gi < LIN) ? xb[gi] : 0.0f;  // zero pad == reference F.pad(0)
    }
    __syncthreads();

    const int wave = tid >> 5;        // wave32
    const int lane = tid & 31;
    const int half = lane >> 4;       // K-half selector for A/B fragments
    const int n    = lane & 15;       // output-position column within tile
    const int t0   = T0 + wave * TILE_N;
    if (t0 >= LOUT) return;           // wave-uniform: EXEC stays all-ones for WMMA

    v8f acc0 = {}, acc1 = {}, acc2 = {}, acc3 = {};

    const int xbase = wave * TILE_N * CSTRIDE + n * CSTRIDE;  // sx index of k=0 for column n
    const int arow  = n;                                      // A: lane%16 = M row in tile

    #pragma unroll
    for (int s = 0; s < 5; ++s) {
        const int k0 = 4 * s + 2 * half;                      // k0 <= 18 always
        // B fragment (4x16, K striped across VGPR+lane-half). No guard needed at k=19:
        // sW column 19 is zero for every row, so that B slot (a staged finite x value)
        // is multiplied by 0 everywhere and contributes nothing.
        v2f bf;
        bf.x = sx[xbase + k0];
        bf.y = sx[xbase + k0 + 1];
        // A fragments for the 4 M-tiles (sW zero-padded at k=19)
        v2f a0, a1, a2, a3;
        a0.x = sW[(0 * 16 + arow) * KPAD + k0];  a0.y = sW[(0 * 16 + arow) * KPAD + k0 + 1];
        a1.x = sW[(1 * 16 + arow) * KPAD + k0];  a1.y = sW[(1 * 16 + arow) * KPAD + k0 + 1];
        a2.x = sW[(2 * 16 + arow) * KPAD + k0];  a2.y = sW[(2 * 16 + arow) * KPAD + k0 + 1];
        a3.x = sW[(3 * 16 + arow) * KPAD + k0];  a3.y = sW[(3 * 16 + arow) * KPAD + k0 + 1];

        acc0 = __builtin_amdgcn_wmma_f32_16x16x4_f32(false, a0, false, bf, (short)0, acc0, false, false);
        acc1 = __builtin_amdgcn_wmma_f32_16x16x4_f32(false, a1, false, bf, (short)0, acc1, false, false);
        acc2 = __builtin_amdgcn_wmma_f32_16x16x4_f32(false, a2, false, bf, (short)0, acc2, false, false);
        acc3 = __builtin_amdgcn_wmma_f32_16x16x4_f32(false, a3, false, bf, (short)0, acc3, false, false);
    }

    // ---- threshold channel for this lane's column t (zero pad never crosses +/-2.99)
    float mx = -1.0e30f, mn = 1.0e30f;
    #pragma unroll
    for (int k = 0; k < KSZ; ++k) {
        const float v = sx[xbase + k];
        mx = fmaxf(mx, v);
        mn = fminf(mn, v);
    }
    const float thr = (mx > UPPERV) ? 1.0f : ((mn < LOWERV) ? -1.0f : 0.0f);

    // ---- store: C/D layout -> VGPR j holds channel (m*16 + j + 8*half), column n
    const int t = t0 + n;                 // tiles never straddle LOUT (20000 % 16 == 0)
    const int chh = half * 8;
    float* ob = out + (long)bb * OUTC * LOUT + t;
    #pragma unroll
    for (int j = 0; j < 8; ++j) {
        {
            const int ch = 0 * 16 + j + chh;
            float v = acc0[j] + sb[ch];
            if (ch == 1) v = thr;         // overwrite threshold slot
            ob[(long)ch * LOUT] = v;
        }
        {
            const int ch = 1 * 16 + j + chh;
            ob[(long)ch * LOUT] = acc1[j] + sb[ch];
        }
        {
            const int ch = 2 * 16 + j + chh;
            ob[(long)ch * LOUT] = acc2[j] + sb[ch];
        }
        {
            const int ch = 3 * 16 + j + chh;
            ob[(long)ch * LOUT] = acc3[j] + sb[ch];
        }
    }
}

extern "C" void kernel_launch(void* const* d_in, const int* in_sizes, int n_in,
                              void* d_out, int out_size, void* d_ws, size_t ws_size,
                              hipStream_t stream) {
    (void)in_sizes; (void)n_in; (void)out_size; (void)d_ws; (void)ws_size;
    const float* x    = (const float*)d_in[0];
    const float* W    = (const float*)d_in[1];
    const float* bias = (const float*)d_in[2];
    float* out        = (float*)d_out;

    const int B = 64;
    dim3 grid(B * BLOCKS_PER_BATCH);   // 64 batches x 157 blocks (8 wave-tiles of 16 each)
    dim3 block(256);                   // 8 waves (wave32)
    Conv1dWithMeanAndThresholdChannels_kernel<<<grid, block, 0, stream>>>(x, W, bias, out);
}